// BatchedGeometryComputation_16939351015985
// MI455X (gfx1250) — compile-verified
//
#include <hip/hip_runtime.h>
#include <hip/hip_bf16.h>

typedef __attribute__((ext_vector_type(2))) float v2f;
typedef __attribute__((ext_vector_type(8))) float v8f;

// ---------------------------------------------------------------------------
// Kernel 1: segment sums + counts with wave-aggregated global atomics.
// block_id is sorted, so runs of equal ids are contiguous: segmented shuffle
// reduction within each wave32, only run-head lanes issue global_atomic_add_f32
// (~2 atomics/value/wave instead of 32 with ~40 atoms per block).
// ---------------------------------------------------------------------------
__global__ void seg_accum_kernel(const float* __restrict__ pos,
                                 const int* __restrict__ bid,
                                 float* __restrict__ sums,     // [n_blocks*3]
                                 float* __restrict__ counts,   // [n_blocks]
                                 int n_atoms) {
    int gid  = blockIdx.x * blockDim.x + threadIdx.x;
    int lane = threadIdx.x & 31;
    bool valid = gid < n_atoms;

    int   b = valid ? bid[gid] : -1;
    long  p3 = (long)gid * 3;
    float x = valid ? pos[p3 + 0] : 0.0f;
    float y = valid ? pos[p3 + 1] : 0.0f;
    float z = valid ? pos[p3 + 2] : 0.0f;
    float c = valid ? 1.0f : 0.0f;

    // segmented tree reduction over contiguous equal-bid runs
    #pragma unroll
    for (int off = 1; off < 32; off <<= 1) {
        int   ob = __shfl_down(b, off, 32);
        float ox = __shfl_down(x, off, 32);
        float oy = __shfl_down(y, off, 32);
        float oz = __shfl_down(z, off, 32);
        float oc = __shfl_down(c, off, 32);
        if (lane + off < 32 && ob == b) { x += ox; y += oy; z += oz; c += oc; }
    }

    int  pb   = __shfl_up(b, 1, 32);
    bool head = (lane == 0) || (pb != b);
    if (head && b >= 0) {
        atomicAdd(&sums[3 * (long)b + 0], x);
        atomicAdd(&sums[3 * (long)b + 1], y);
        atomicAdd(&sums[3 * (long)b + 2], z);
        atomicAdd(&counts[b], c);
    }
}

// ---------------------------------------------------------------------------
// Kernel 2: centroids = sums / max(counts, 1)   -> output slice 0
// ---------------------------------------------------------------------------
__global__ void centroid_kernel(const float* __restrict__ sums,
                                const float* __restrict__ counts,
                                float* __restrict__ out_centroids,
                                int n_blocks) {
    int i = blockIdx.x * blockDim.x + threadIdx.x;
    if (i < n_blocks) {
        float cnt = fmaxf(counts[i], 1.0f);
        out_centroids[3 * (long)i + 0] = sums[3 * (long)i + 0] / cnt;
        out_centroids[3 * (long)i + 1] = sums[3 * (long)i + 1] / cnt;
        out_centroids[3 * (long)i + 2] = sums[3 * (long)i + 2] / cnt;
    }
}

// ---------------------------------------------------------------------------
// Kernel 3: per-atom rel/dist + RBF via V_WMMA_F32_16X16X4_F32.
// exponent[m][j] = [d^2, d, 1, 0] @ B ; B folds 1/(2w^2), centers and log2(e),
// so rbf = exp2(D) — one v_exp_f32 per element after the WMMA.
// Each wave processes 32 atoms = two 16x16 WMMA tiles (16 atoms x 16 centers).
// Stores use one base pointer + constant immediate offsets; big streaming
// outputs are non-temporal so L2 stays free for the centroid table and pos.
// ---------------------------------------------------------------------------
__global__ void geom_rbf_kernel(const float* __restrict__ pos,
                                const int* __restrict__ bid,
                                const float* __restrict__ centroids,
                                const float* __restrict__ centers,
                                const float* __restrict__ widths,
                                float* __restrict__ out_rel,
                                float* __restrict__ out_dist,
                                float* __restrict__ out_rbf,
                                int n_atoms) {
    int gid  = blockIdx.x * blockDim.x + threadIdx.x;
    int lane = threadIdx.x & 31;
    bool valid = gid < n_atoms;
    int  a = valid ? gid : (n_atoms - 1);   // clamp: keep loads safe, EXEC uniform

    int   b  = bid[a];
    long  p3 = (long)a * 3;
    float px = pos[p3 + 0], py = pos[p3 + 1], pz = pos[p3 + 2];
    float cx = centroids[3 * (long)b + 0];
    float cy = centroids[3 * (long)b + 1];
    float cz = centroids[3 * (long)b + 2];
    float rx = px - cx, ry = py - cy, rz = pz - cz;
    float d  = sqrtf(rx * rx + ry * ry + rz * rz);

    if (valid) {
        __builtin_nontemporal_store(rx, &out_rel[p3 + 0]);
        __builtin_nontemporal_store(ry, &out_rel[p3 + 1]);
        __builtin_nontemporal_store(rz, &out_rel[p3 + 2]);
        __builtin_nontemporal_store(d,  &out_dist[a]);
    }

    // ---- B matrix (4x16), log2(e) folded in ----
    int   j   = lane & 15;
    float cj  = centers[j];
    float wj  = widths[j];
    float k   = 0.5f / (wj * wj);
    const float LOG2E = 1.4426950408889634f;
    float B0 = -k * LOG2E;                 // coeff of d^2
    float B1 =  2.0f * cj * k * LOG2E;     // coeff of d
    float B2 = -cj * cj * k * LOG2E;       // constant
    bool hi = lane >= 16;
    v2f Bm;
    Bm.x = hi ? B2 : B0;   // lanes 0-15: K=0 row ; lanes 16-31: K=2 row
    Bm.y = hi ? 0.0f : B1; // lanes 0-15: K=1 row ; lanes 16-31: K=3 row (0)

    long tile_base  = (long)blockIdx.x * blockDim.x + (threadIdx.x & ~31);
    bool wave_full  = (tile_base + 32) <= (long)n_atoms;   // wave-uniform
    float d_sw = __shfl_xor(d, 16, 32);    // distances of the other half-wave

    // one base pointer per lane; every store is a constant immediate offset:
    //   tile t, row r  ->  byte offset t*1024 + r*64
    float* rbase = out_rbf + ((size_t)tile_base + (hi ? 8u : 0u)) * 16u + (unsigned)j;

    #pragma unroll
    for (int t = 0; t < 2; ++t) {
        float dt = (t == 0) ? d : d_sw;    // tile t atoms live in lanes 0-15 (+16t)
        v2f A;
        A.x = hi ? 1.0f : dt * dt;         // K=0 (d^2) | K=2 (1)
        A.y = hi ? 0.0f : dt;              // K=1 (d)   | K=3 (0)
        v8f C = {};
        v8f D = __builtin_amdgcn_wmma_f32_16x16x4_f32(
                    false, A, false, Bm, (short)0, C, false, false);

        if (wave_full) {                   // uniform branch: the hot path
            #pragma unroll
            for (int r = 0; r < 8; ++r)
                __builtin_nontemporal_store(__builtin_amdgcn_exp2f(D[r]),
                                            &rbase[(t * 16 + r) * 16]);
        } else {                           // tail waves only (never taken at 4M)
            long abase = tile_base + t * 16 + (hi ? 8 : 0);
            #pragma unroll
            for (int r = 0; r < 8; ++r) {
                if (abase + r < n_atoms)
                    __builtin_nontemporal_store(__builtin_amdgcn_exp2f(D[r]),
                                                &rbase[(t * 16 + r) * 16]);
            }
        }
    }
}

// ---------------------------------------------------------------------------
extern "C" void kernel_launch(void* const* d_in, const int* in_sizes, int n_in,
                              void* d_out, int out_size, void* d_ws, size_t ws_size,
                              hipStream_t stream) {
    const float* pos     = (const float*)d_in[0];
    const int*   bid     = (const int*)d_in[1];
    // d_in[2] = n_blocks device scalar (value derived on host below)
    const float* centers = (const float*)d_in[3];
    const float* widths  = (const float*)d_in[4];

    int n_atoms = in_sizes[1];                               // block_id count
    // out = 3*n_blocks + 3*n_atoms + n_atoms + 16*n_atoms
    int n_blocks = (int)(((long)out_size - 20L * n_atoms) / 3L);

    float* out_c    = (float*)d_out;
    float* out_rel  = out_c   + (size_t)3 * n_blocks;
    float* out_dist = out_rel + (size_t)3 * n_atoms;
    float* out_rbf  = out_dist + (size_t)n_atoms;

    float* sums   = (float*)d_ws;                 // [3*n_blocks]
    float* counts = sums + (size_t)3 * n_blocks;  // [n_blocks]

    hipMemsetAsync(d_ws, 0, (size_t)4 * n_blocks * sizeof(float), stream);

    const int T = 256;
    int gA = (n_atoms + T - 1) / T;
    int gB = (n_blocks + T - 1) / T;

    seg_accum_kernel<<<gA, T, 0, stream>>>(pos, bid, sums, counts, n_atoms);
    centroid_kernel<<<gB, T, 0, stream>>>(sums, counts, out_c, n_blocks);
    geom_rbf_kernel<<<gA, T, 0, stream>>>(pos, bid, out_c, centers, widths,
                                          out_rel, out_dist, out_rbf, n_atoms);
}